// Attention_47914655154335
// MI455X (gfx1250) — compile-verified
//
#include <hip/hip_runtime.h>
#include <hip/hip_bf16.h>

typedef float v2f __attribute__((ext_vector_type(2)));
typedef float v8f __attribute__((ext_vector_type(8)));

#define HIDDEN 256
#define QH 256
#define KH 256
#define BATCH 32
#define NQ 64
#define NK 256

#if defined(__has_builtin)
#if __has_builtin(__builtin_amdgcn_global_load_async_to_lds_b128) && \
    __has_builtin(__builtin_amdgcn_global_load_async_to_lds_b32)
#define USE_ASYNC_LDS 1
#else
#define USE_ASYNC_LDS 0
#endif
#else
#define USE_ASYNC_LDS 0
#endif

#if defined(__has_builtin) && __has_builtin(__builtin_amdgcn_s_wait_asynccnt)
#define WAIT_ASYNC0() __builtin_amdgcn_s_wait_asynccnt(0)
#else
#define WAIT_ASYNC0() asm volatile("s_wait_asynccnt 0x0" ::: "memory")
#endif

// Types matching the builtin signatures reported by the frontend:
//   b128: (int __vector(4) AS1*, int __vector(4) AS3*, imm, imm)
//   b32 : (int AS1*,             int AS3*,             imm, imm)
typedef int vi4 __attribute__((__vector_size__(16)));
#define AS_G __attribute__((address_space(1)))
#define AS_L __attribute__((address_space(3)))

// ---------------------------------------------------------------------------
// Phase 1: batched GEMM  H[b] (MxN) = Wsub (MxK) * X[b] (KxN)  via
// V_WMMA_F32_16X16X4_F32.  One wave (32 threads) computes one 16x16 tile.
//   A frag (v2f): lane<16 -> (M=lane,  K=k0+0 / k0+1)
//                 lane>=16-> (M=lane-16,K=k0+2 / k0+3)
//   B frag (v2f): mirrored over N
//   C/D  (v8f):  VGPR r -> M = r + 8*(lane>>4), N = lane&15
// ---------------------------------------------------------------------------
__global__ __launch_bounds__(32) void gemm_wmma_f32(
    const float* __restrict__ W,   // (M x ldw) row-major; use cols [colOff, colOff+K)
    int ldw, int colOff,
    const float* __restrict__ X,   // batched (K x ldx) row-major
    int ldx, long xBatchStride,
    float* __restrict__ H,         // batched (M x N) row-major
    int N, long hBatchStride, int K)
{
    const int lane   = threadIdx.x & 31;
    const int halfId = lane >> 4;
    const int lm     = lane & 15;
    const int n0     = blockIdx.x * 16;
    const int m0     = blockIdx.y * 16;
    const int b      = blockIdx.z;

    const float* Xb = X + (long)b * xBatchStride;
    const float* Wr = W + (long)(m0 + lm) * ldw + colOff;

    v8f acc = {};
    for (int k0 = 0; k0 < K; k0 += 4) {
        const int ka = k0 + 2 * halfId;
        v2f a;
        a.x = Wr[ka];
        a.y = Wr[ka + 1];
        v2f bf;
        bf.x = Xb[(long)ka * ldx + n0 + lm];
        bf.y = Xb[(long)(ka + 1) * ldx + n0 + lm];
        // 8 args: (neg_a, A, neg_b, B, c_mod, C, reuse_a, reuse_b)
        acc = __builtin_amdgcn_wmma_f32_16x16x4_f32(
            false, a, false, bf, (short)0, acc, false, false);
    }

    float* Hb = H + (long)b * hBatchStride;
#pragma unroll
    for (int r = 0; r < 8; ++r) {
        const int row = r + 8 * halfId;
        Hb[(long)(m0 + row) * N + n0 + lm] = acc[r];
    }
}

// ---------------------------------------------------------------------------
// Phase 2: attns[b,n,m] = sum_h v[h] * tanh(hq[b,h,n] + hk[b,h,m])
// Block: 256 threads (8 waves). Wave w owns n = n0 + w (hq read is a
// conflict-free LDS broadcast); lane l owns m = l + 32*j, j=0..7 (8 indep
// accumulator chains keep the trans pipe fed). h chunked by 32 through a
// DOUBLE-BUFFERED LDS stage filled by async global->LDS copies (ASYNCcnt).
// ---------------------------------------------------------------------------
__global__ __launch_bounds__(256) void combine_tanh(
    const float* __restrict__ hq,   // (B, HIDDEN, NQ)
    const float* __restrict__ hk,   // (B, HIDDEN, NK)
    const float* __restrict__ v,    // (HIDDEN)
    float* __restrict__ out)        // (B, NQ*NK)
{
    __shared__ float sh_hk[2][32 * NK];   // 2 x 32 KB
    __shared__ float sh_hq[2][32 * 8];
    __shared__ float sh_v[2][32];

    const int t    = threadIdx.x;
    const int w    = t >> 5;
    const int lane = t & 31;
    const int n0   = blockIdx.x * 8;
    const int b    = blockIdx.y;

    const float* hqB = hq + (long)b * HIDDEN * NQ;
    const float* hkB = hk + (long)b * HIDDEN * NK;

    // Stage one 32-h chunk into LDS buffer `buf`.
    auto stage = [&](int buf, int h0) {
#if USE_ASYNC_LDS
        // hk chunk: 32 x 256 floats = 2048 float4, 8 async b128 per thread
#pragma unroll
        for (int i = 0; i < 8; ++i) {
            const int idx4 = t + 256 * i;          // float4 index
            const int hl   = idx4 >> 6;            // 64 float4 per h-row
            const int m4   = idx4 & 63;
            const float* gp = hkB + (long)(h0 + hl) * NK + m4 * 4;
            __builtin_amdgcn_global_load_async_to_lds_b128(
                (AS_G vi4*)gp, (AS_L vi4*)&sh_hk[buf][idx4 * 4], 0, 0);
        }
        {
            const int hl = t >> 3, nl = t & 7;
            const float* gp = hqB + (long)(h0 + hl) * NQ + n0 + nl;
            __builtin_amdgcn_global_load_async_to_lds_b32(
                (AS_G int*)gp, (AS_L int*)&sh_hq[buf][t], 0, 0);
        }
        if (t < 32) {
            __builtin_amdgcn_global_load_async_to_lds_b32(
                (AS_G int*)(v + h0 + t), (AS_L int*)&sh_v[buf][t], 0, 0);
        }
#else
#pragma unroll
        for (int i = 0; i < 32; ++i) {
            const int idx = t + 256 * i;
            const int hl  = idx >> 8;
            const int m   = idx & 255;
            sh_hk[buf][idx] = hkB[(long)(h0 + hl) * NK + m];
        }
        {
            const int hl = t >> 3, nl = t & 7;
            sh_hq[buf][t] = hqB[(long)(h0 + hl) * NQ + n0 + nl];
        }
        if (t < 32) sh_v[buf][t] = v[h0 + t];
#endif
    };

    float acc[8];
#pragma unroll
    for (int j = 0; j < 8; ++j) acc[j] = 0.0f;

    stage(0, 0);   // prefetch chunk 0

    for (int c = 0; c < HIDDEN / 32; ++c) {
        WAIT_ASYNC0();       // this wave's outstanding async copies done
        __syncthreads();     // everyone's copies visible; prev compute done

        if (c + 1 < HIDDEN / 32) stage((c + 1) & 1, (c + 1) * 32);

        const float* __restrict__ hkS = sh_hk[c & 1];
        const float* __restrict__ hqS = sh_hq[c & 1];
        const float* __restrict__ vS  = sh_v[c & 1];

        for (int hl = 0; hl < 32; ++hl) {
            const float hqv = hqS[hl * 8 + w];   // broadcast within wave
            const float vh  = vS[hl];
#pragma unroll
            for (int j = 0; j < 8; ++j) {
                float x = hqv + hkS[hl * NK + lane + 32 * j];
                float th;
                asm("v_tanh_f32 %0, %1" : "=v"(th) : "v"(x));
                acc[j] = fmaf(vh, th, acc[j]);
            }
        }
    }

    const int n = n0 + w;
    float* ob = out + (long)b * (NQ * NK) + (long)n * NK;
#pragma unroll
    for (int j = 0; j < 8; ++j) ob[lane + 32 * j] = acc[j];
}

// ---------------------------------------------------------------------------
extern "C" void kernel_launch(void* const* d_in, const int* in_sizes, int n_in,
                              void* d_out, int out_size, void* d_ws, size_t ws_size,
                              hipStream_t stream) {
    const float* att_query = (const float*)d_in[0];  // (32, 256, 64)
    const float* att_key   = (const float*)d_in[1];  // (32, 256, 256)
    const float* v         = (const float*)d_in[2];  // (256)
    const float* W         = (const float*)d_in[3];  // (256, 512)
    float* out = (float*)d_out;

    float* hq = (float*)d_ws;                        // 32*256*64  = 524288 f
    float* hk = hq + (long)BATCH * HIDDEN * NQ;      // 32*256*256 = 2097152 f

    // hq = Wq * att_query : M=256, N=64, K=256
    gemm_wmma_f32<<<dim3(NQ / 16, HIDDEN / 16, BATCH), 32, 0, stream>>>(
        W, QH + KH, /*colOff=*/0,
        att_query, NQ, (long)HIDDEN * NQ,
        hq, NQ, (long)HIDDEN * NQ, HIDDEN);

    // hk = Wk * att_key : M=256, N=256, K=256
    gemm_wmma_f32<<<dim3(NK / 16, HIDDEN / 16, BATCH), 32, 0, stream>>>(
        W, QH + KH, /*colOff=*/QH,
        att_key, NK, (long)HIDDEN * NK,
        hk, NK, (long)HIDDEN * NK, HIDDEN);

    // combine: grid (NQ/8, B) = (8, 32), 256 threads
    combine_tanh<<<dim3(NQ / 8, BATCH), 256, 0, stream>>>(hq, hk, v, out);
}